// HybridANFIS_51719996178808
// MI455X (gfx1250) — compile-verified
//
#include <hip/hip_runtime.h>
#include <hip/hip_bf16.h>

// Problem constants (HybridANFIS): B=2048, I=16, M=4, R=4096, C=10
#define BATCH 2048
#define NIN   16
#define NMF   4
#define NRULE 4096
#define NOUT  10
#define JC    170           // (I+1)*C
#define JCP   176           // padded to 11 WMMA N-tiles
#define NTILES_N (JCP / 16) // 11
#define KGRP  (NRULE / 4)   // 1024 k-groups of 4

typedef __attribute__((ext_vector_type(2))) float v2f;
typedef __attribute__((ext_vector_type(8))) float v8f;

// ---------------------------------------------------------------------------
// Kernel 0: prep
//  - g[b,i,m]   = -(x-c)^2/(2w^2) + 1e-9             (B*64 values)
//  - x_ext[b,j] = j<16 ? x[b,j] : 1                   (B*17, direct to d_out)
//  - packed[r]  = 16 x 2-bit MF indices in one u32    (R values)
//  - Bt         = consequents pre-swizzled into the exact per-lane WMMA
//                 B-operand layout (zero-padded cols 170..175), so the GEMM
//                 inner loop does ONE coalesced b64 load per lane, no branches.
//                 Bt[((nt*KGRP + kg)*32 + lane)] = { B[4kg+2h][n], B[4kg+2h+1][n] }
//                 with n = nt*16 + (lane&15), h = lane>>4.
// ---------------------------------------------------------------------------
__global__ void __launch_bounds__(256)
prep_kernel(const float* __restrict__ x,
            const float* __restrict__ centers,
            const float* __restrict__ widths,
            const float* __restrict__ cons,
            const int*   __restrict__ rule_idx,
            float*       __restrict__ g,
            float*       __restrict__ xext,
            unsigned*    __restrict__ packed,
            float*       __restrict__ Bt)
{
    int tid = blockIdx.x * blockDim.x + threadIdx.x;

    if (tid < BATCH * NIN * NMF) {
        int b  = tid >> 6;
        int im = tid & 63;
        int i  = im >> 2;
        int m  = im & 3;
        float d = x[b * NIN + i] - centers[i * NMF + m];
        float w = widths[i * NMF + m];
        g[tid] = -(d * d) / (2.0f * w * w) + 1e-9f;
    }

    if (tid < BATCH * (NIN + 1)) {
        int b = tid / (NIN + 1);
        int j = tid % (NIN + 1);
        xext[tid] = (j < NIN) ? x[b * NIN + j] : 1.0f;
    }

    if (tid < NRULE) {
        unsigned code = 0;
        #pragma unroll
        for (int i = 0; i < NIN; ++i) {
            unsigned idx = (unsigned)rule_idx[i * NRULE + tid] & 3u;
            code |= idx << (2 * i);
        }
        packed[tid] = code;
    }

    if (tid < NTILES_N * KGRP * 32) {
        int nt   = tid / (KGRP * 32);
        int rem  = tid - nt * (KGRP * 32);
        int kg   = rem >> 5;
        int lane = rem & 31;
        int row  = lane & 15;
        int half = lane >> 4;
        int col  = nt * 16 + row;
        int k0   = 4 * kg + 2 * half;
        v2f v;
        if (col < JC) {
            v.x = cons[(size_t)k0 * JC + col];
            v.y = cons[(size_t)(k0 + 1) * JC + col];
        } else {
            v.x = 0.0f;
            v.y = 0.0f;
        }
        *(v2f*)(Bt + (size_t)tid * 2) = v;
    }
}

// ---------------------------------------------------------------------------
// Kernel 1: firing + normalize.  One block (8 wave32) per batch row.
//  firing[r] = exp( sum_i g[b, i, idx(i,r)] )   via packed 2-bit indices
//  norm_fs written coalesced to d_out.
// ---------------------------------------------------------------------------
__global__ void __launch_bounds__(256)
firing_kernel(const float*    __restrict__ g,
              const unsigned* __restrict__ packed,
              float*          __restrict__ norm)
{
    __shared__ unsigned sCodes[NRULE];
    __shared__ float    sFire[NRULE];
    __shared__ float    sG[NIN * NMF];
    __shared__ float    sRed[256];

    const int b = blockIdx.x;
    const int t = threadIdx.x;

    #pragma unroll
    for (int c = 0; c < NRULE / 256; ++c)
        sCodes[c * 256 + t] = packed[c * 256 + t];
    if (t < NIN * NMF)
        sG[t] = g[b * (NIN * NMF) + t];
    __syncthreads();

    float lsum = 0.0f;
    #pragma unroll
    for (int c = 0; c < NRULE / 256; ++c) {
        int r = c * 256 + t;
        unsigned code = sCodes[r];
        float s = 0.0f;
        #pragma unroll
        for (int i = 0; i < NIN; ++i)
            s += sG[i * 4 + ((code >> (2 * i)) & 3u)];
        float f = __expf(s);
        sFire[r] = f;
        lsum += f;
    }

    sRed[t] = lsum;
    __syncthreads();
    #pragma unroll
    for (int off = 128; off > 0; off >>= 1) {
        if (t < off) sRed[t] += sRed[t + off];
        __syncthreads();
    }
    const float inv = 1.0f / (sRed[0] + 1e-9f);

    float* dst = norm + (size_t)b * NRULE;
    #pragma unroll
    for (int c = 0; c < NRULE / 256; ++c) {
        int r = c * 256 + t;
        dst[r] = sFire[r] * inv;
    }
}

// ---------------------------------------------------------------------------
// Kernel 2: T[b, jc] = sum_r norm_fs[b,r] * cons[r, jc]   (2048x4096 @ 4096x176)
// Full-precision f32 WMMA 16x16x4. One 16x16 tile per wave; 8 waves/block.
// A layout (16x4 f32): lane<16 -> {K,K+1}, lane>=16 -> {K+2,K+3}, row = lane&15
// B operand read pre-swizzled from Bt: one coalesced b64 per lane per step.
// C/D layout: VGPR r -> M = r + 8*(lane>=16), N = lane&15.
// ---------------------------------------------------------------------------
__global__ void __launch_bounds__(256)
wmma_kernel(const float* __restrict__ A,    // norm_fs  B x NRULE
            const float* __restrict__ Bt,   // swizzled B operand
            float*       __restrict__ T)    // B x JCP
{
    const int wave = threadIdx.x >> 5;
    const int lane = threadIdx.x & 31;
    const int tile = blockIdx.x * 8 + wave;
    const int mt = tile / NTILES_N;
    const int nt = tile % NTILES_N;

    const int row  = lane & 15;
    const int half = lane >> 4;

    const float* aPtr = A + (size_t)(mt * 16 + row) * NRULE + half * 2;
    const v2f*   bPtr = (const v2f*)Bt + (size_t)nt * KGRP * 32 + lane;

    v8f acc = {};
    #pragma unroll 8
    for (int kg = 0; kg < KGRP; ++kg) {
        v2f a = *(const v2f*)(aPtr + kg * 4);
        v2f b = bPtr[(size_t)kg * 32];
        acc = __builtin_amdgcn_wmma_f32_16x16x4_f32(
                  false, a, false, b, (short)0, acc, false, false);
    }

    float* dst = T + (size_t)(mt * 16 + half * 8) * JCP + nt * 16 + row;
    #pragma unroll
    for (int r = 0; r < 8; ++r)
        dst[(size_t)r * JCP] = acc[r];
}

// ---------------------------------------------------------------------------
// Kernel 3: y_hat[b,c] = sum_j x_ext[b,j] * T[b, j*C + c]
// ---------------------------------------------------------------------------
__global__ void __launch_bounds__(256)
final_kernel(const float* __restrict__ T,
             const float* __restrict__ xext,
             float*       __restrict__ y)
{
    int tid = blockIdx.x * blockDim.x + threadIdx.x;
    if (tid >= BATCH * NOUT) return;
    int b = tid / NOUT;
    int c = tid % NOUT;
    const float* tp = T + (size_t)b * JCP;
    const float* xe = xext + b * (NIN + 1);
    float s = 0.0f;
    #pragma unroll
    for (int j = 0; j < NIN + 1; ++j)
        s += xe[j] * tp[j * NOUT + c];
    y[tid] = s;
}

// ---------------------------------------------------------------------------
extern "C" void kernel_launch(void* const* d_in, const int* in_sizes, int n_in,
                              void* d_out, int out_size, void* d_ws, size_t ws_size,
                              hipStream_t stream)
{
    const float* x        = (const float*)d_in[0];   // B x I
    const float* centers  = (const float*)d_in[1];   // I x M
    const float* widths   = (const float*)d_in[2];   // I x M
    const float* cons     = (const float*)d_in[3];   // R x (I+1) x C
    const int*   rule_idx = (const int*)d_in[4];     // I x R

    float* out   = (float*)d_out;
    float* yhat  = out;                                   // B*C
    float* norm  = out + (size_t)BATCH * NOUT;            // B*R
    float* xext  = norm + (size_t)BATCH * NRULE;          // B*(I+1)

    char* ws = (char*)d_ws;
    float*    g      = (float*)ws;                        // B*64 f32      = 512 KB
    unsigned* packed = (unsigned*)(ws + 524288);          // R u32         =  16 KB
    float*    T      = (float*)(ws + 540672);             // B*176 f32     = 1.44 MB
    float*    Bt     = (float*)(ws + 1982464);            // 11*1024*64 f32= 2.88 MB

    // K0: memberships (log-domain), x_ext, packed rule codes, swizzled B operand
    {
        int nthreads = NTILES_N * KGRP * 32;              // 360448 (largest job)
        prep_kernel<<<dim3((nthreads + 255) / 256), dim3(256), 0, stream>>>(
            x, centers, widths, cons, rule_idx, g, xext, packed, Bt);
    }

    // K1: firing + normalization, one block per batch row
    firing_kernel<<<dim3(BATCH), dim3(256), 0, stream>>>(g, packed, norm);

    // K2: norm_fs @ cons  (f32 WMMA), 128 x 11 tiles, 8 waves/block
    wmma_kernel<<<dim3((128 * NTILES_N) / 8), dim3(256), 0, stream>>>(
        norm, Bt, T);

    // K3: contract with x_ext -> y_hat
    final_kernel<<<dim3((BATCH * NOUT + 255) / 256), dim3(256), 0, stream>>>(
        T, xext, yhat);
}